// Mask_9131100471519
// MI455X (gfx1250) — compile-verified
//
#include <hip/hip_runtime.h>

// MAE masking: B=32, N=1024, D=768, mask_ratio=0.75 -> K=768 masked, U=256 unmasked.
#define MAE_B 32
#define MAE_N 1024
#define MAE_D 768
#define MAE_K 768
#define MAE_U 256

// Raw clang ext-vector (NOT HIP_vector_type) so nontemporal builtins accept it.
typedef float v4f __attribute__((ext_vector_type(4)));

// Flat float offsets of the 5 concatenated outputs (return order).
static constexpr size_t O1 = 0;                                   // with_mask            (B,N,D)
static constexpr size_t O2 = O1 + (size_t)MAE_B * MAE_N * MAE_D;  // unmasked_patches     (B,U,D)
static constexpr size_t O3 = O2 + (size_t)MAE_B * MAE_U * MAE_D;  // bool_mask_expanded   (B,N,D)
static constexpr size_t O4 = O3 + (size_t)MAE_B * MAE_N * MAE_D;  // masked_indices       (B,K)
static constexpr size_t O5 = O4 + (size_t)MAE_B * MAE_K;          // unmasked_indices     (B,U)

// ---------------------------------------------------------------------------
// Kernel 1: per-batch rank/scatter/partition. One 1024-thread block per batch.
//   rank(i)  = #{ j : noise[j] < noise[i]  or (noise[j]==noise[i] && j<i) }
//   mask[rank(i)] = (i < K)            (== perm < K since perm[rank(i)] = i)
//   stable partition via prefix sum of mask -> masked/unmasked index lists.
// Also seeds mask scalar into bool_mask_expanded[b, j, 0] for later kernels.
// ---------------------------------------------------------------------------
__global__ __launch_bounds__(1024) void mae_rank_kernel(const float* __restrict__ noise,
                                                        float* __restrict__ out) {
  __shared__ float s_noise[MAE_N];
  __shared__ int   s_mask[MAE_N];
  const int b   = blockIdx.x;
  const int tid = threadIdx.x;

  s_noise[tid] = noise[(size_t)b * MAE_N + tid];
  __syncthreads();

  const float x = s_noise[tid];
  int rank = 0;
#pragma unroll 8
  for (int j = 0; j < MAE_N; ++j) {
    const float v = s_noise[j];
    rank += (int)((v < x) || ((v == x) && (j < tid)));  // stable tie-break
  }
  s_mask[rank] = (tid < MAE_K) ? 1 : 0;  // scatter: bool_mask[rank(tid)] = tid<K
  __syncthreads();

  const int m = s_mask[tid];
  int prefTrue = 0;  // #True strictly before tid (uniform LDS broadcast reads)
  for (int j = 0; j < tid; ++j) prefTrue += s_mask[j];

  if (m) {
    out[O4 + (size_t)b * MAE_K + prefTrue] = (float)tid;            // masked_indices
  } else {
    out[O5 + (size_t)b * MAE_U + (tid - prefTrue)] = (float)tid;    // unmasked_indices
  }
  // Seed per-row mask scalar; kernel 2 broadcasts it across D (incl. rewriting slot 0).
  out[O3 + (size_t)(b * MAE_N + tid) * MAE_D] = (float)m;
}

// ---------------------------------------------------------------------------
// Kernel 2: one 192-thread block per row (192 * float4 = 768 floats).
//   with_mask[row]          = m ? mask_emb : patch[row]   (masked rows: no patch read)
//   bool_mask_expanded[row] = m broadcast
// Branch is row-uniform -> no divergence; all traffic is 128-bit. Output
// streams are write-once -> non-temporal stores (TH=NT) so the 216 MB of
// output doesn't evict `patch` (96 MB, fits in the 192 MB L2) before the
// kernel-3 gather re-reads the unmasked rows.
// ---------------------------------------------------------------------------
__global__ __launch_bounds__(192) void mae_fill_kernel(const float* __restrict__ patch,
                                                       const float* __restrict__ mask_emb,
                                                       float* __restrict__ out) {
  const int row = blockIdx.x;   // 0 .. B*N-1
  const int t   = threadIdx.x;  // 0 .. 191
  __shared__ float s_m;
  if (t == 0) s_m = out[O3 + (size_t)row * MAE_D];
  __syncthreads();
  const float  mf   = s_m;
  const size_t base = (size_t)row * MAE_D + 4 * (size_t)t;

  v4f v;
  if (mf != 0.0f) {
    v = *reinterpret_cast<const v4f*>(mask_emb + 4 * t);  // hot in L2/L0
  } else {
    v = *reinterpret_cast<const v4f*>(patch + base);
  }
  const v4f vm = {mf, mf, mf, mf};
  __builtin_nontemporal_store(v, reinterpret_cast<v4f*>(out + O1 + base));
  __builtin_nontemporal_store(vm, reinterpret_cast<v4f*>(out + O3 + base));
}

// ---------------------------------------------------------------------------
// Kernel 3: gather the 256 unmasked rows per batch. One wave32 per row,
// 6 x b128 per lane. Preferred path: CDNA5 async global->LDS->global DMA
// (ASYNCcnt-tracked, no VGPR round-trip); fallback: plain b128 copies.
//
// Probe-confirmed builtin params (round 1 diagnostic):
//   (v4i __device__* /*AS1*/, v4i __shared__* /*AS3*/, imm int offset, imm int cpol)
// where v4i = int __attribute__((ext_vector_type(4))).
// ---------------------------------------------------------------------------
#if defined(__gfx1250__) &&                                                \
    __has_builtin(__builtin_amdgcn_global_load_async_to_lds_b128) &&       \
    __has_builtin(__builtin_amdgcn_global_store_async_from_lds_b128) &&    \
    __has_builtin(__builtin_amdgcn_s_wait_asynccnt)
#define MAE_ASYNC_LDS 1
typedef int v4i __attribute__((ext_vector_type(4)));
typedef v4i __attribute__((address_space(1))) *gv4i_p;  // global (AS1) int4*
typedef v4i __attribute__((address_space(3))) *lv4i_p;  // LDS    (AS3) int4*
#else
#define MAE_ASYNC_LDS 0
#endif

__global__ __launch_bounds__(256) void mae_gather_kernel(const float* __restrict__ patch,
                                                         float* __restrict__ out) {
  const int wave = threadIdx.x >> 5;
  const int lane = threadIdx.x & 31;
  const int row  = blockIdx.x * 8 + wave;  // 0 .. B*U-1
  const int b    = row >> 8;               // row / U
  const int s    = row & (MAE_U - 1);

  const int idx = (int)out[O5 + (size_t)b * MAE_U + s];  // exact small float
  const float* src = patch + ((size_t)b * MAE_N + (size_t)idx) * MAE_D;
  float*       dst = out + O2 + (size_t)row * MAE_D;

#if MAE_ASYNC_LDS
  __shared__ v4f s_buf[8][6 * 32];  // 24 KB: one 3 KB row slab per wave
  v4f* lbase = &s_buf[wave][0];
#pragma unroll
  for (int c = 0; c < 6; ++c) {
    const int chunk = lane + 32 * c;
    __builtin_amdgcn_global_load_async_to_lds_b128(
        (gv4i_p)(src + 4 * chunk), (lv4i_p)(lbase + chunk), 0, 0);
  }
  __builtin_amdgcn_s_wait_asynccnt(0);  // LDS writes visible before store-from-LDS reads
#pragma unroll
  for (int c = 0; c < 6; ++c) {
    const int chunk = lane + 32 * c;
    __builtin_amdgcn_global_store_async_from_lds_b128(
        (gv4i_p)(dst + 4 * chunk), (lv4i_p)(lbase + chunk), 0, 0);
  }
  __builtin_amdgcn_s_wait_asynccnt(0);
#else
#pragma unroll
  for (int c = 0; c < 6; ++c) {
    const int chunk = lane + 32 * c;
    const v4f v = *reinterpret_cast<const v4f*>(src + 4 * chunk);
    __builtin_nontemporal_store(v, reinterpret_cast<v4f*>(dst + 4 * chunk));
  }
#endif
}

extern "C" void kernel_launch(void* const* d_in, const int* in_sizes, int n_in,
                              void* d_out, int out_size, void* d_ws, size_t ws_size,
                              hipStream_t stream) {
  (void)in_sizes; (void)n_in; (void)out_size; (void)d_ws; (void)ws_size;
  const float* patch    = (const float*)d_in[0];  // (B,N,D) f32
  const float* noise    = (const float*)d_in[1];  // (B,N)   f32
  const float* mask_emb = (const float*)d_in[2];  // (D,)    f32
  float* out = (float*)d_out;

  mae_rank_kernel<<<MAE_B, MAE_N, 0, stream>>>(noise, out);
  mae_fill_kernel<<<MAE_B * MAE_N, MAE_D / 4, 0, stream>>>(patch, mask_emb, out);
  mae_gather_kernel<<<(MAE_B * MAE_U) / 8, 256, 0, stream>>>(patch, out);
}